// ROIPooling1_38001870635744
// MI455X (gfx1250) — compile-verified
//
#include <hip/hip_runtime.h>
#include <hip/hip_bf16.h>

// ---------------------------------------------------------------------------
// ROI max-pooling for MI455X (gfx1250).
//   features: (B=2, C=128, H=50, W=50) f32
//   rois:     (B=2, R=64, 4)           f32   [px, py, qx, qy] * (1/SCALE)
//   out:      (B, R, C, 7, 7)          f32
// Strategy: one workgroup per (b, r, channel-chunk). Stage the ROI rectangle
// for CH=4 channels into LDS with CDNA5 async global->LDS copies (ASYNCcnt /
// s_wait_asynccnt), then each thread max-reduces one output bin from LDS.
// This op is a gather+max (no contraction dimension) -> WMMA does not apply;
// the gfx1250-specific win is the async global->LDS data path.
// ---------------------------------------------------------------------------

#define POOL   7
#define B_N    2
#define C_N    128
#define H_N    50
#define W_N    50
#define R_N    64
#define CH     4            // channels per workgroup
#define NTHREADS 256        // 8 wave32 waves
#define TILE_STRIDE 2504    // >= 50*50, padded a bit

static_assert(C_N % CH == 0, "channel chunking");

#if defined(__has_builtin)
#  if __has_builtin(__builtin_amdgcn_global_load_async_to_lds_b32) && \
      __has_builtin(__builtin_amdgcn_s_wait_asynccnt)
#    define USE_ASYNC_LDS 1
#  endif
#endif
#ifndef USE_ASYNC_LDS
#  define USE_ASYNC_LDS 0
#endif

typedef __attribute__((address_space(1))) int* gint_ptr;  // global (device) int*
typedef __attribute__((address_space(3))) int* lint_ptr;  // LDS int*

__global__ __launch_bounds__(NTHREADS)
void roi_pool_gfx1250_kernel(const float* __restrict__ feat,
                             const float* __restrict__ rois,
                             float* __restrict__ out)
{
    __shared__ float tile[CH * TILE_STRIDE];   // 40,064 B static LDS

    const int br  = blockIdx.x;          // b*R + r
    const int b   = br / R_N;
    const int c0  = blockIdx.y * CH;
    const int tid = threadIdx.x;

    // ---- ROI integer parameters (replicated per thread; scalar-friendly) ----
    const float scale = 1.0f / 16.0f;
    const float* roi = rois + (size_t)br * 4;
    int px = (int)rintf(roi[0] * scale);
    int py = (int)rintf(roi[1] * scale);
    int qx = (int)rintf(roi[2] * scale);
    int qy = (int)rintf(roi[3] * scale);
    px = px < 0 ? 0 : (px > W_N - 1 ? W_N - 1 : px);
    py = py < 0 ? 0 : (py > H_N - 1 ? H_N - 1 : py);
    qx = qx < 0 ? 0 : (qx > W_N - 1 ? W_N - 1 : qx);
    qy = qy < 0 ? 0 : (qy > H_N - 1 ? H_N - 1 : qy);

    const int roi_w = qx - px + 1;                 // >= 1
    const int roi_h = qy - py + 1;                 // >= 1
    int ps_w = (roi_w + POOL - 1) / POOL; if (ps_w < 1) ps_w = 1;
    int ps_h = (roi_h + POOL - 1) / POOL; if (ps_h < 1) ps_h = 1;
    const int pad_l = (ps_w * POOL - roi_w) >> 1;  // non-negative
    const int pad_t = (ps_h * POOL - roi_h) >> 1;

    // ---- Stage CH x roi_h x roi_w rectangle into LDS ------------------------
    const int hw    = roi_h * roi_w;
    const int total = CH * hw;
    const float* fbase = feat + ((size_t)b * C_N + c0) * (H_N * W_N);

    for (int e = tid; e < total; e += NTHREADS) {
        const int ch  = e / hw;
        const int rem = e - ch * hw;
        const int row = rem / roi_w;
        const int col = rem - row * roi_w;
        const float* src = fbase + (size_t)ch * (H_N * W_N)
                         + (size_t)(py + row) * W_N + (px + col);
        float* dst = &tile[ch * TILE_STRIDE + rem];
#if USE_ASYNC_LDS
        __builtin_amdgcn_global_load_async_to_lds_b32(
            (gint_ptr)(const_cast<float*>(src)),
            (lint_ptr)(dst),
            /*offset=*/0, /*cpol=*/0);
#else
        *dst = *src;
#endif
    }
#if USE_ASYNC_LDS
    __builtin_amdgcn_s_wait_asynccnt(0);   // drain this wave's async copies
#endif
    __syncthreads();                       // make LDS visible to all waves

    // ---- Each thread computes one (ch, ih, iw) bin max ----------------------
    const int nout = CH * POOL * POOL;     // 196 outputs per workgroup
    for (int o = tid; o < nout; o += NTHREADS) {
        const int ch  = o / (POOL * POOL);
        const int rem = o - ch * (POOL * POOL);
        const int ih  = rem / POOL;
        const int iw  = rem - ih * POOL;

        const float* tch = &tile[ch * TILE_STRIDE];
        const int rbase = ih * ps_h - pad_t;
        const int cbase = iw * ps_w - pad_l;

        float m = -INFINITY;   // in-bin set is non-empty -> result finite
        for (int kh = 0; kh < ps_h; ++kh) {
            const int rr = rbase + kh;
            const bool vr = (rr >= 0) & (rr < roi_h);
            for (int kw = 0; kw < ps_w; ++kw) {
                const int cc = cbase + kw;
                const bool v = vr & (cc >= 0) & (cc < roi_w);
                const int idx = v ? (rr * roi_w + cc) : 0;  // safe LDS index
                const float val = v ? tch[idx] : 0.0f;      // padding -> 0.0
                m = fmaxf(m, val);
            }
        }
        // out[b, r, c0+ch, ih, iw]
        out[(((size_t)br * C_N + (c0 + ch)) * POOL + ih) * POOL + iw] = m;
    }
}

extern "C" void kernel_launch(void* const* d_in, const int* in_sizes, int n_in,
                              void* d_out, int out_size, void* d_ws, size_t ws_size,
                              hipStream_t stream)
{
    (void)in_sizes; (void)n_in; (void)out_size; (void)d_ws; (void)ws_size;
    const float* feat = (const float*)d_in[0];   // (2,128,50,50) f32
    const float* rois = (const float*)d_in[1];   // (2,64,4)      f32
    float* out = (float*)d_out;                  // (2,64,128,7,7) f32

    dim3 grid(B_N * R_N, C_N / CH);   // (128, 32)
    dim3 block(NTHREADS);
    roi_pool_gfx1250_kernel<<<grid, block, 0, stream>>>(feat, rois, out);
}